// BatteryCell_56813827391770
// MI455X (gfx1250) — compile-verified
//
#include <hip/hip_runtime.h>
#include <math.h>

// ---------------------------------------------------------------------------
// Battery-cell Euler step, B = 2M independent cells. Memory-bound streaming
// kernel (~80 B/cell -> HBM roofline ~7.2us at 23.3 TB/s).
// CDNA5 paths: GLOBAL_LOAD_ASYNC_TO_LDS_B128 (ASYNCcnt) staging of the state
// vectors + s_wait_asynccnt; native TRANS-pipe v_tanh/v_log/v_sqrt/v_rcp to
// keep the VALU stream far below the bandwidth-matched ops budget.
// ---------------------------------------------------------------------------

#define BLOCK 256

// ---- TRANS-pipe helpers ---------------------------------------------------
__device__ __forceinline__ float fast_tanh(float x) {
#if __has_builtin(__builtin_amdgcn_tanhf)
    return __builtin_amdgcn_tanhf(x);                 // v_tanh_f32
#else
    // tanh(x) = 1 - 2/(e^{2x}+1); exp2/rcp are single TRANS ops.
    float ex = __builtin_amdgcn_exp2f(x * 2.88539008177792681472f); // 2*log2(e)
    return 1.0f - 2.0f * __builtin_amdgcn_rcpf(ex + 1.0f);
#endif
}

__device__ __forceinline__ float fast_ln(float x) {   // natural log, x > 0
    return __builtin_amdgcn_logf(x) * 0.69314718055994530942f; // v_log_f32 * ln2
}

__device__ __forceinline__ float fast_rcp(float x) {
    return __builtin_amdgcn_rcpf(x);                  // v_rcp_f32 (~1 ulp)
}

__device__ __forceinline__ float fast_asinh(float x) {
    float t = fabsf(x);
    // Overflow-safe: for huge t, asinh(t) = ln(2t); else ln(t + sqrt(t^2+1)).
    float s   = __builtin_amdgcn_sqrtf(fmaf(t, t, 1.0f));
    float arg = (t > 1.0e18f) ? (2.0f * t) : (t + s);
    return copysignf(fast_ln(arg), x);
}

__global__ __launch_bounds__(BLOCK) void battery_step_kernel(
    const float* __restrict__ inputs,   // (B,)
    const float* __restrict__ states,   // (B,8) row-major
    const float* __restrict__ qMaxA,    // (B,)
    const float* __restrict__ RoA,      // (B,)
    const float* __restrict__ tDiffA,   // (1,)
    const float* __restrict__ Wp1,      // (1,8)
    const float* __restrict__ bp1,      // (8,)
    const float* __restrict__ Wp2,      // (8,4) row-major
    const float* __restrict__ bp2,      // (4,)
    const float* __restrict__ Wp3,      // (4,1)
    const float* __restrict__ bp3,      // (1,)
    const float* __restrict__ Wn,       // (1,1)
    const float* __restrict__ bn,       // (1,)
    float* __restrict__ outV,           // (B,)
    float* __restrict__ outX,           // (B,8)
    int n)
{
    __shared__ float4 tile[BLOCK * 2];  // 32 B per thread (8 floats)

    const int tid = threadIdx.x;
    const unsigned e = blockIdx.x * BLOCK + tid;      // B*8 < 2^31: 32-bit ok
    if ((int)e >= n) return;

    // ---- Stage this thread's 8 state floats: global -> LDS (async path) ---
    const float* gp = states + (size_t)e * 8;         // 32B-aligned
#if defined(__gfx1250__)
    {
        unsigned lds_off = (unsigned)(size_t)(&tile[tid * 2]);
        asm volatile(
            "global_load_async_to_lds_b128 %0, %1, off\n\t"
            "global_load_async_to_lds_b128 %0, %1, off offset:16"
            :
            : "v"(lds_off), "v"(gp)
            : "memory");
        asm volatile("s_wait_asynccnt 0" ::: "memory");
    }
#else
    tile[tid * 2 + 0] = ((const float4*)gp)[0];
    tile[tid * 2 + 1] = ((const float4*)gp)[1];
#endif

    // Per-cell scalars, loaded after the async LDS staging has completed.
    const float i_in  = inputs[e];
    const float qMax  = qMaxA[e];
    const float Ro    = RoA[e];
    const float rtD   = fast_rcp(tDiffA[0]);          // 1/tDiffusion

    const float4 s0 = tile[tid * 2 + 0];              // ds_load_b128
    const float4 s1 = tile[tid * 2 + 1];

    const float Tb  = s0.x, Vo  = s0.y, Vsn = s0.z, Vsp = s0.w;
    const float qnB = s1.x, qnS = s1.y, qpB = s1.z, qpS = s1.w;

    // ---- Constants (denominators folded into reciprocals) -----------------
    const float R_GAS     = 8.3144621f;
    const float Fc        = 96487.0f;
    const float INV_VOLB  = 1.0f / 1.98e-05f;         // 1/(VOL-VOLS)
    const float INV_VOLS  = 1.0f / 2.2e-06f;
    const float INV_SN    = 5000.0f;                  // 1/2e-4
    const float INV_SP    = 5000.0f;
    const float KN        = 20000.0f, KP = 20000.0f;
    const float INV_TO    = 0.1f;
    const float INV_TS    = 1.0f / 90.0f;             // TSN == TSP
    const float U0P       = 4.03f, U0N = 0.01f;
    const float RTF_ALPHA = (float)(8.3144621 / 96487.0 / 0.5); // R/F/alpha
    const float RTF       = (float)(8.3144621 / 96487.0);       // R/F

    const float qSMax = qMax * 10000.0f * 2.2e-06f * (1.0f / 2.2e-05f);
    const float rqS   = fast_rcp(qSMax);

    float xpS = fminf(fmaxf(qpS * rqS, 1e-18f), 1.0f);
    float xnS = fminf(fmaxf(qnS * rqS, 1e-18f), 1.0f);

    // (1-x)^0.5 * x^0.5
    float Jn0 = fmaf(KN, __builtin_amdgcn_sqrtf(1.0f - xnS) * __builtin_amdgcn_sqrtf(xnS), 1e-18f);
    float Jp0 = fmaf(KP, __builtin_amdgcn_sqrtf(1.0f - xpS) * __builtin_amdgcn_sqrtf(xpS), 1e-18f);

    float qdotDiffBSn = (qnB * INV_VOLB - qnS * INV_VOLS) * rtD;
    float qdotDiffBSp = (qpB * INV_VOLB - qpS * INV_VOLS) * rtD;

    float Vodot = (i_in * Ro * 10.0f - Vo) * INV_TO;

    float kT = Tb * RTF_ALPHA;                        // R*Tb/F/alpha
    float VsnNom = kT * fast_asinh(i_in * INV_SN * (0.5f * fast_rcp(Jn0)));
    float VspNom = kT * fast_asinh(i_in * INV_SP * (0.5f * fast_rcp(Jp0)));
    float Vsndot = (VsnNom - Vsn) * INV_TS;
    float Vspdot = (VspNom - Vsp) * INV_TS;

    // ---- Euler update (DT = 1) --------------------------------------------
    const float X0 = Tb;
    const float X1 = Vo  + Vodot;
    const float X2 = Vsn + Vsndot;
    const float X3 = Vsp + Vspdot;
    const float X4 = qnB - qdotDiffBSn;
    const float X5 = qnS + (qdotDiffBSn - i_in);
    const float X6 = qpB - qdotDiffBSp;
    const float X7 = qpS + (i_in + qdotDiffBSp);

    const float xpS2 = X7 * rqS;
    const float xnS2 = X5 * rqS;

    // ---- Shared-weight MLP 1 -> 8 -> 4 -> 1 (weights scalarized) ----------
    float h1[8];
#pragma unroll
    for (int j = 0; j < 8; ++j)
        h1[j] = fast_tanh(fmaf(xpS2, Wp1[j], bp1[j]));

    float h2[4];
#pragma unroll
    for (int k = 0; k < 4; ++k) {
        float acc = bp2[k];
#pragma unroll
        for (int j = 0; j < 8; ++j)
            acc = fmaf(h1[j], Wp2[j * 4 + k], acc);
        h2[k] = fast_tanh(acc);
    }

    float VepMLP = bp3[0];
#pragma unroll
    for (int k = 0; k < 4; ++k)
        VepMLP = fmaf(h2[k], Wp3[k], VepMLP);

    const float VenMLP = fmaf(xnS2, Wn[0], bn[0]);

    // ---- Open-circuit potentials ------------------------------------------
    float lp  = fminf(fmaxf((1.0f - xpS2) * fast_rcp(xpS2), 1e-18f), 1e18f);
    float ln_ = fminf(fmaxf((1.0f - xnS2) * fast_rcp(xnS2), 1e-18f), 1e18f);

    const float rt  = X0 * RTF;
    const float Vep = U0P + rt * fast_ln(lp)  + VepMLP;
    const float Ven = U0N + rt * fast_ln(ln_) + VenMLP;
    const float V   = Vep - Ven - X1 - X2 - X3;

    // ---- Stores: V (b32) + Xnew (2x b128), fully coalesced ----------------
    outV[e] = V;
    float4* po = (float4*)(outX + (size_t)e * 8);
    float4 o0; o0.x = X0; o0.y = X1; o0.z = X2; o0.w = X3;
    float4 o1; o1.x = X4; o1.y = X5; o1.z = X6; o1.w = X7;
    po[0] = o0;
    po[1] = o1;
}

extern "C" void kernel_launch(void* const* d_in, const int* in_sizes, int n_in,
                              void* d_out, int out_size, void* d_ws, size_t ws_size,
                              hipStream_t stream) {
    const float* inputs = (const float*)d_in[0];
    const float* states = (const float*)d_in[1];
    const float* qMax   = (const float*)d_in[2];
    const float* Ro     = (const float*)d_in[3];
    const float* tDiff  = (const float*)d_in[4];
    const float* Wp1    = (const float*)d_in[5];
    const float* bp1    = (const float*)d_in[6];
    const float* Wp2    = (const float*)d_in[7];
    const float* bp2    = (const float*)d_in[8];
    const float* Wp3    = (const float*)d_in[9];
    const float* bp3    = (const float*)d_in[10];
    const float* Wn     = (const float*)d_in[11];
    const float* bn     = (const float*)d_in[12];

    const int n = in_sizes[0];               // B = 2097152
    float* outV = (float*)d_out;             // (B,)
    float* outX = outV + (size_t)n;          // (B,8)

    const int blocks = (n + BLOCK - 1) / BLOCK;
    hipLaunchKernelGGL(battery_step_kernel, dim3(blocks), dim3(BLOCK), 0, stream,
                       inputs, states, qMax, Ro, tDiff,
                       Wp1, bp1, Wp2, bp2, Wp3, bp3, Wn, bn,
                       outV, outX, n);
}